// BaseSOM_51496657879311
// MI455X (gfx1250) — compile-verified
//
#include <hip/hip_runtime.h>

typedef __attribute__((ext_vector_type(2))) float v2f;
typedef __attribute__((ext_vector_type(8))) float v8f;

#define DIM 512
#define MN  4096
#define LDA 516  // padded LDS row stride (floats): +4 pad -> conflict-free b64 frag reads

// ---------------------------------------------------------------------------
// Kernel 0: w2[n] = sum_k weights[n][k]^2   (one wave per row)
// ---------------------------------------------------------------------------
__global__ __launch_bounds__(256) void som_w2_kernel(const float* __restrict__ weights,
                                                     float* __restrict__ w2) {
  const int wave = threadIdx.x >> 5;
  const int lane = threadIdx.x & 31;
  const int row  = blockIdx.x * 8 + wave;
  const float4* wr = (const float4*)(weights + (size_t)row * DIM);
  float acc = 0.0f;
#pragma unroll
  for (int i = 0; i < 4; ++i) {
    float4 q = wr[lane + 32 * i];
    acc += q.x * q.x + q.y * q.y + q.z * q.z + q.w * q.w;
  }
#pragma unroll
  for (int m = 16; m >= 1; m >>= 1) acc += __shfl_xor(acc, m, 32);
  if (lane == 0) w2[row] = acc;
}

// ---------------------------------------------------------------------------
// Kernel 1: fused fp32 WMMA GEMM (batch @ weights^T) + per-row argmax of
//           score = 2*dot - w2   (== argmin of squared euclidean distance)
// Block: 128 threads (4 waves). Block bid owns batch rows [bid*16, bid*16+16).
// Wave w handles N-tiles nt = w, w+4, ..., accumulating K=512 via 128
// v_wmma_f32_16x16x4_f32 ops per tile (4 rotating accumulator chains).
// K-loop unroll capped at 2 (8 WMMA / iter) to stay under 256 VGPRs: avoids
// s_set_vgpr_msb toggling on the WMMA issue path and keeps occupancy high.
// ---------------------------------------------------------------------------
__global__ __launch_bounds__(128) void som_bmu_kernel(const float* __restrict__ batch,
                                                      const float* __restrict__ weights,
                                                      const float* __restrict__ w2,
                                                      int* __restrict__ bmu) {
  __shared__ __align__(16) float As[16 * LDA];
  __shared__ float cV[4][16];
  __shared__ int   cI[4][16];

  const int tid  = threadIdx.x;
  const int lane = tid & 31;
  const int wv   = tid >> 5;
  const int bid  = blockIdx.x;  // 256 blocks -> 256 M-tiles of 16 batch rows

  // Cooperative load of the 16x512 A tile into padded LDS (float4, coalesced).
  const float4* bp = (const float4*)(batch + (size_t)bid * 16 * DIM);
#pragma unroll
  for (int i = 0; i < 16; ++i) {
    int idx = tid + 128 * i;          // 0..2047 float4s
    int row = idx >> 7;               // /128 float4 per row
    int c4  = idx & 127;
    float4 q = bp[idx];
    *(float4*)&As[row * LDA + c4 * 4] = q;
  }
  __syncthreads();

  // Fragment addressing for V_WMMA_F32_16X16X4_F32 (wave32):
  //   A: lane l, vgpr r -> A[l%16][k0 + 2*(l/16) + r]  (contiguous float2)
  //   B: lane l, vgpr r -> B[k0 + 2*(l/16) + r][l%16]  == weights[n][k] (same pattern)
  const int half = lane >> 4;   // 0 or 1
  const int l16  = lane & 15;
  const float* aRow = &As[l16 * LDA + 2 * half];

  float bestV[8];
  int   bestI[8];
#pragma unroll
  for (int r = 0; r < 8; ++r) { bestV[r] = -3.402823466e38f; bestI[r] = 0; }

  for (int nt = wv; nt < MN / 16; nt += 4) {
    const int n = nt * 16 + l16;
    const float* wRow = weights + (size_t)n * DIM + 2 * half;
    v8f acc0 = {}, acc1 = {}, acc2 = {}, acc3 = {};
#pragma unroll 2
    for (int ks = 0; ks < DIM / 4; ks += 4) {  // 4 K-steps (16 K) per iteration
      v2f a0 = *(const v2f*)(aRow + 4 * ks);
      v2f b0 = *(const v2f*)(wRow + 4 * ks);
      acc0 = __builtin_amdgcn_wmma_f32_16x16x4_f32(false, a0, false, b0,
                                                   (short)0, acc0, false, false);
      v2f a1 = *(const v2f*)(aRow + 4 * ks + 4);
      v2f b1 = *(const v2f*)(wRow + 4 * ks + 4);
      acc1 = __builtin_amdgcn_wmma_f32_16x16x4_f32(false, a1, false, b1,
                                                   (short)0, acc1, false, false);
      v2f a2 = *(const v2f*)(aRow + 4 * ks + 8);
      v2f b2 = *(const v2f*)(wRow + 4 * ks + 8);
      acc2 = __builtin_amdgcn_wmma_f32_16x16x4_f32(false, a2, false, b2,
                                                   (short)0, acc2, false, false);
      v2f a3 = *(const v2f*)(aRow + 4 * ks + 12);
      v2f b3 = *(const v2f*)(wRow + 4 * ks + 12);
      acc3 = __builtin_amdgcn_wmma_f32_16x16x4_f32(false, a3, false, b3,
                                                   (short)0, acc3, false, false);
    }
    const float w2n = w2[n];
#pragma unroll
    for (int r = 0; r < 8; ++r) {
      // C layout: vgpr r, lanes 0-15 -> row r, col lane; lanes 16-31 -> row 8+r
      float s = 2.0f * (acc0[r] + acc1[r] + acc2[r] + acc3[r]) - w2n;
      if (s > bestV[r]) { bestV[r] = s; bestI[r] = n; }  // strict '>' keeps lowest n
    }
  }

  // Reduce across the 16 lanes of each half-wave (xor masks 8..1 stay in-half).
#pragma unroll
  for (int r = 0; r < 8; ++r) {
    float v = bestV[r];
    int   ix = bestI[r];
#pragma unroll
    for (int m = 8; m >= 1; m >>= 1) {
      float ov = __shfl_xor(v, m, 32);
      int   oi = __shfl_xor(ix, m, 32);
      if (ov > v || (ov == v && oi < ix)) { v = ov; ix = oi; }
    }
    if (lane == 0)  { cV[wv][r]     = v; cI[wv][r]     = ix; }  // rows 0..7
    if (lane == 16) { cV[wv][8 + r] = v; cI[wv][8 + r] = ix; }  // rows 8..15
  }
  __syncthreads();

  // Combine the 4 waves' disjoint N subsets (tie -> lower index).
  if (tid < 16) {
    float v = cV[0][tid];
    int   ix = cI[0][tid];
#pragma unroll
    for (int w = 1; w < 4; ++w) {
      float ov = cV[w][tid];
      int   oi = cI[w][tid];
      if (ov > v || (ov == v && oi < ix)) { v = ov; ix = oi; }
    }
    bmu[bid * 16 + tid] = ix;
  }
}

// ---------------------------------------------------------------------------
// Kernel 2: out[row][j] = exp(-((ji-bi)^2+(jj-bj)^2)/sigma^2)
//         = efi[ji] * efj[jj]   (separable -> 128 exps per row, then stores)
// One 128-thread block per batch row; fully coalesced 64MB write (HBM floor).
// ---------------------------------------------------------------------------
__global__ __launch_bounds__(128) void som_out_kernel(const int* __restrict__ bmu,
                                                      const int* __restrict__ it_p,
                                                      const int* __restrict__ decay_p,
                                                      float* __restrict__ out) {
  __shared__ float efi[64];
  __shared__ float efj[64];
  const int row = blockIdx.x;
  const int t   = threadIdx.x;
  const int idx = bmu[row];
  const int bi = idx >> 6;
  const int bj = idx & 63;
  const float lr  = __expf(-(float)it_p[0] / (float)decay_p[0]);
  const float sig = 32.0f * lr;  // SIGMA = max(M,N)/2 = 32
  const float inv = 1.0f / (sig * sig);
  if (t < 64) {
    int d = t - bi;
    efi[t] = __expf(-(float)(d * d) * inv);
  } else {
    int c = t - 64;
    int d = c - bj;
    efj[c] = __expf(-(float)(d * d) * inv);
  }
  __syncthreads();
  float* op = out + (size_t)row * MN;
  const float fj = efj[t & 63];  // (t + 128k) & 63 == t & 63
#pragma unroll
  for (int k = 0; k < 32; ++k) {
    int j = t + (k << 7);
    op[j] = efi[j >> 6] * fj;  // efi read is an LDS broadcast per half-wave
  }
}

// ---------------------------------------------------------------------------
extern "C" void kernel_launch(void* const* d_in, const int* in_sizes, int n_in,
                              void* d_out, int out_size, void* d_ws, size_t ws_size,
                              hipStream_t stream) {
  const float* batch   = (const float*)d_in[0];  // [4096, 512] fp32
  const float* weights = (const float*)d_in[1];  // [4096, 512] fp32
  // d_in[2] = locations (unused: it is the analytic 64x64 meshgrid)
  const int* decay = (const int*)d_in[3];        // scalar int
  const int* it    = (const int*)d_in[4];        // scalar int

  float* w2  = (float*)d_ws;                             // 4096 floats
  int*   bmu = (int*)((char*)d_ws + MN * sizeof(float)); // 4096 ints
  float* out = (float*)d_out;                            // [4096, 4096] fp32

  som_w2_kernel <<<MN / 8, 256, 0, stream>>>(weights, w2);
  som_bmu_kernel<<<MN / 16, 128, 0, stream>>>(batch, weights, w2, bmu);
  som_out_kernel<<<4096, 128, 0, stream>>>(bmu, it, decay, out);
}